// SoftDiceLoss_57380763075170
// MI455X (gfx1250) — compile-verified
//
#include <hip/hip_runtime.h>

// SoftDiceLoss on MI455X (gfx1250, wave32).
// Memory-bound streaming reduction: 134 MB read once -> one scalar.
// Big sums ride the tensor pipe via chained V_WMMA_F32_16X16X4_F32 with an
// all-ones B (D += A x 1 == row-sum accumulate, exact fp32); predicates/max
// ride the VALU; finish with wave shuffles + LDS + per-sample atomics.

typedef float v2f __attribute__((ext_vector_type(2)));
typedef float v4f __attribute__((ext_vector_type(4)));
typedef float v8f __attribute__((ext_vector_type(8)));

namespace {
constexpr int kB               = 64;          // batch
constexpr int kHW              = 512 * 512;   // C*H*W with C == 1
constexpr int kBlocksPerSample = 32;
constexpr int kThreads         = 256;         // 8 wave32 waves
constexpr int kV4PerSample     = kHW / 4;                      // 65536
constexpr int kV4PerBlock      = kV4PerSample / kBlocksPerSample; // 2048
constexpr int kIters           = kV4PerBlock / kThreads;       // 8
constexpr int kAcc             = 6; // sum_p, sum_t, sum_pt, corr1, cntSR, maxT(bits)
}

__global__ void sdl_init(float* __restrict__ ws) {
  ws[threadIdx.x] = 0.0f;   // launched with kB*kAcc threads
}

__device__ __forceinline__ v8f wmma_rowsum(v2f a, v2f ones, v8f acc) {
  // D = A(16x4) * Ones(4x16) + C  ->  every D[m][n] accumulates rowsum(m).
  return __builtin_amdgcn_wmma_f32_16x16x4_f32(
      /*neg_a=*/false, a, /*neg_b=*/false, ones,
      /*c_mod=*/(short)0, acc, /*reuse_a=*/false, /*reuse_b=*/false);
}

__global__ void __launch_bounds__(kThreads)
sdl_partial(const float* __restrict__ p, const float* __restrict__ t,
            float* __restrict__ ws) {
  const int sample = blockIdx.x / kBlocksPerSample;
  const int blk    = blockIdx.x - sample * kBlocksPerSample;
  const size_t base = (size_t)sample * kHW + (size_t)blk * (kV4PerBlock * 4);
  const v4f* __restrict__ p4 = (const v4f*)(p + base);
  const v4f* __restrict__ t4 = (const v4f*)(t + base);
  const int tid = threadIdx.x;

  v8f accP = {}, accT = {}, accPT = {};
  const v2f ones = {1.0f, 1.0f};
  float corr1 = 0.0f, cntSR = 0.0f, maxT = 0.0f;

#pragma unroll
  for (int i = 0; i < kIters; ++i) {
    const int idx = i * kThreads + tid;
    v4f pv = __builtin_nontemporal_load(&p4[idx]);   // b128, TH=NT stream
    v4f tv = __builtin_nontemporal_load(&t4[idx]);
    if (i + 1 < kIters) {                            // global_prefetch_b8 ahead
      __builtin_prefetch(&p4[idx + kThreads], 0, 0);
      __builtin_prefetch(&t4[idx + kThreads], 0, 0);
    }

    // Tensor-pipe row-sum accumulation (2 VGPRs of A per WMMA = 64 floats/wave).
    v2f pa = {pv.x, pv.y}, pb = {pv.z, pv.w};
    v2f ta = {tv.x, tv.y}, tb = {tv.z, tv.w};
    v2f ma = {pv.x * tv.x, pv.y * tv.y}, mb = {pv.z * tv.z, pv.w * tv.w};
    accP  = wmma_rowsum(pa, ones, accP);
    accP  = wmma_rowsum(pb, ones, accP);
    accT  = wmma_rowsum(ta, ones, accT);
    accT  = wmma_rowsum(tb, ones, accT);
    accPT = wmma_rowsum(ma, ones, accPT);
    accPT = wmma_rowsum(mb, ones, accPT);

    // VALU stream: predicate counts + max (targets are exactly {0,1}).
#pragma unroll
    for (int k = 0; k < 4; ++k) {
      const float pe = pv[k], te = tv[k];
      const bool SR  = pe > 0.5f;       // probs > 0.5
      const bool GT1 = te > 0.5f;       // t == 1  (candidate GT when max==1)
      corr1 += (SR == GT1) ? 1.0f : 0.0f;
      cntSR += SR ? 1.0f : 0.0f;        // corr when max==0 (GT all-true)
      maxT = fmaxf(maxT, te);
    }
  }

  // Per-wave reduce. D-layout: lane n holds rows 0-7 of column n (lanes 0-15)
  // or rows 8-15 (lanes 16-31); every column equals the full sum, so one
  // horizontal add + shfl_xor(16) gives the wave total in every lane.
  float sp = 0.0f, st = 0.0f, spt = 0.0f;
#pragma unroll
  for (int k = 0; k < 8; ++k) { sp += accP[k]; st += accT[k]; spt += accPT[k]; }
  sp  += __shfl_xor(sp, 16);
  st  += __shfl_xor(st, 16);
  spt += __shfl_xor(spt, 16);
#pragma unroll
  for (int off = 16; off > 0; off >>= 1) {
    corr1 += __shfl_xor(corr1, off);
    cntSR += __shfl_xor(cntSR, off);
    maxT   = fmaxf(maxT, __shfl_xor(maxT, off));
  }

  __shared__ float red[kThreads / 32][kAcc];
  const int wave = tid >> 5;
  if ((tid & 31) == 0) {
    red[wave][0] = sp;    red[wave][1] = st;    red[wave][2] = spt;
    red[wave][3] = corr1; red[wave][4] = cntSR; red[wave][5] = maxT;
  }
  __syncthreads();

  if (tid < kAcc) {
    float v = red[0][tid];
    if (tid == 5) {
#pragma unroll
      for (int w = 1; w < kThreads / 32; ++w) v = fmaxf(v, red[w][5]);
      // non-negative floats: uint compare == float compare
      atomicMax((unsigned int*)&ws[sample * kAcc + 5], __float_as_uint(v));
    } else {
#pragma unroll
      for (int w = 1; w < kThreads / 32; ++w) v += red[w][tid];
      atomicAdd(&ws[sample * kAcc + tid], v);
    }
  }
}

__global__ void sdl_finalize(const float* __restrict__ ws, float* __restrict__ out) {
  // 64 threads, one per sample; 2-wave reduce to the mean.
  const int b = threadIdx.x;
  const float* s = ws + b * kAcc;
  const float sp    = s[0];
  const float st    = s[1];
  const float spt   = s[2];
  const float corr1 = s[3];
  const float cntSR = s[4];
  const float maxT  = __uint_as_float(((const unsigned int*)s)[5]);

  const float corr = (maxT >= 0.5f) ? corr1 : cntSR;
  const float acc  = corr;                         // corr / C with C == 1
  const float dice = 2.0f * (spt + 1.0f) / (sp + st + 1.0f);  // SMOOTH = 1
  const float score = (acc == 1.0f) ? 1.0f : dice;
  float term = 1.0f - score;

#pragma unroll
  for (int off = 16; off > 0; off >>= 1) term += __shfl_xor(term, off);
  __shared__ float sred[2];
  if ((threadIdx.x & 31) == 0) sred[threadIdx.x >> 5] = term;
  __syncthreads();
  if (threadIdx.x == 0) out[0] = (sred[0] + sred[1]) / (float)kB;
}

extern "C" void kernel_launch(void* const* d_in, const int* in_sizes, int n_in,
                              void* d_out, int out_size, void* d_ws, size_t ws_size,
                              hipStream_t stream) {
  const float* probs   = (const float*)d_in[0];
  const float* targets = (const float*)d_in[1];
  float* out = (float*)d_out;
  float* ws  = (float*)d_ws;   // kB * kAcc floats of accumulators

  sdl_init<<<1, kB * kAcc, 0, stream>>>(ws);
  sdl_partial<<<kB * kBlocksPerSample, kThreads, 0, stream>>>(probs, targets, ws);
  sdl_finalize<<<1, kB, 0, stream>>>(ws, out);
}